// G2EquivariantFeedForward_43224550866984
// MI455X (gfx1250) — compile-verified
//
#include <hip/hip_runtime.h>
#include <math.h>

typedef float v2f __attribute__((ext_vector_type(2)));
typedef float v4f __attribute__((ext_vector_type(4)));
typedef float v8f __attribute__((ext_vector_type(8)));

struct Quat { float w, x, y, z; };

__device__ __forceinline__ Quat qmul(Quat a, Quat b) {
    Quat r;
    r.w = a.w*b.w - a.x*b.x - a.y*b.y - a.z*b.z;
    r.x = a.w*b.x + a.x*b.w + a.y*b.z - a.z*b.y;
    r.y = a.w*b.y - a.x*b.z + a.y*b.w + a.z*b.x;
    r.z = a.w*b.z + a.x*b.y - a.y*b.x + a.z*b.w;
    return r;
}
__device__ __forceinline__ Quat qconj(Quat q) { return Quat{q.w, -q.x, -q.y, -q.z}; }
__device__ __forceinline__ Quat qadd(Quat a, Quat b) { return Quat{a.w+b.w, a.x+b.x, a.y+b.y, a.z+b.z}; }
__device__ __forceinline__ Quat qsub(Quat a, Quat b) { return Quat{a.w-b.w, a.x-b.x, a.y-b.y, a.z-b.z}; }

struct Oct { Quat a, b; };
// _oct_mul(o1,o2): re = qm(a,c) - qm(conj(d), b) ; im = qm(d,a) + qm(b, conj(c))
__device__ __forceinline__ Oct omul(Oct p, Oct q) {
    Oct r;
    r.a = qsub(qmul(p.a, q.a), qmul(qconj(q.b), p.b));
    r.b = qadd(qmul(q.b, p.a), qmul(p.b, qconj(q.a)));
    return r;
}

// Branch-free erf (Abramowitz-Stegun 7.1.26, |err| <= 1.5e-7), exact-GELU wrapper.
// Avoids libm erff's exec-divergent dual-path expansion.
__device__ __forceinline__ float gelu_exact(float x) {
    const float z  = x * 0.70710678118654752f;   // x / sqrt(2)
    const float az = fabsf(z);
    const float t  = __builtin_amdgcn_rcpf(fmaf(0.3275911f, az, 1.0f));
    float p = fmaf(1.061405429f, t, -1.453152027f);
    p = fmaf(p, t, 1.421413741f);
    p = fmaf(p, t, -0.284496736f);
    p = fmaf(p, t, 0.254829592f);
    p = p * t;
    const float e    = __builtin_amdgcn_exp2f(az * az * -1.4426950408889634f);
    const float erfa = fmaf(-p, e, 1.0f);        // erf(|z|)
    const float erfz = copysignf(erfa, z);
    return 0.5f * x * (1.0f + erfz);
}

__global__ __launch_bounds__(256)
void g2ff_kernel(const float* __restrict__ o,
                 const float* __restrict__ W1,
                 const float* __restrict__ b1,
                 const float* __restrict__ W2,
                 const float* __restrict__ b2,
                 const float* __restrict__ alpha,
                 float* __restrict__ out,
                 int n_oct, int stride)
{
    constexpr int ITERS = 4;

    const int lane = threadIdx.x & 31;
    const bool lo  = lane < 16;          // lower half of the wave
    const int half = lo ? 0 : 8;         // M-offset of this half in D/C layout
    const int m    = lane & 15;

    // ---- hoisted weight setup (per wave, reused across ITERS octonion groups) ----
    // A = 16x4 tile of W1 (transposed layer-1: hT = W1 * invT). K=2,3 rows are zero.
    v2f amat[2];
#pragma unroll
    for (int t = 0; t < 2; ++t) {
        amat[t].x = lo ? W1[(m + 16*t)*2 + 0] : 0.0f;   // K=0 (lanes<16) / K=2 pad
        amat[t].y = lo ? W1[(m + 16*t)*2 + 1] : 0.0f;   // K=1 (lanes<16) / K=3 pad
    }
    // C = bias b1 broadcast in 16x16 f32 C/D layout (constant per-M row).
    v8f cbias[2];
#pragma unroll
    for (int t = 0; t < 2; ++t)
#pragma unroll
        for (int v = 0; v < 8; ++v)
            cbias[t][v] = b1[v + half + 16*t];
    // Layer-2 coefficients matched to the D layout (M = v + half + 16t).
    float w2c[3][2][8];
#pragma unroll
    for (int c = 0; c < 3; ++c)
#pragma unroll
        for (int t = 0; t < 2; ++t)
#pragma unroll
            for (int v = 0; v < 8; ++v)
                w2c[c][t][v] = W2[c*32 + v + half + 16*t];

    const float bb0 = b2[0], bb1 = b2[1], bb2v = b2[2];
    const float lam = 1.0f / (1.0f + expf(-alpha[0]));
    const float oml = 1.0f - lam;

    const int gtid = blockIdx.x * blockDim.x + threadIdx.x;

#pragma unroll
    for (int it = 0; it < ITERS; ++it) {
        int oct = gtid + it * stride;
        oct = oct < n_oct ? oct : n_oct - 1;   // clamp (keeps EXEC all-ones for WMMA)
        const long base = (long)oct * 2;

        const v4f* op = (const v4f*)o + base;
        v4f q0 = __builtin_nontemporal_load(op);       // streamed once: NT hint
        v4f q1 = __builtin_nontemporal_load(op + 1);
        Oct oh;
        oh.a = Quat{q0.x, q0.y, q0.z, q0.w};
        oh.b = Quat{q1.x, q1.y, q1.z, q1.w};

        const float rinv = q0.x;
        float nsq = q0.y*q0.y;
        nsq = fmaf(q0.z, q0.z, nsq); nsq = fmaf(q0.w, q0.w, nsq);
        nsq = fmaf(q1.x, q1.x, nsq); nsq = fmaf(q1.y, q1.y, nsq);
        nsq = fmaf(q1.z, q1.z, nsq); nsq = fmaf(q1.w, q1.w, nsq);
        const float nrm = __builtin_amdgcn_sqrtf(nsq);

        // ---- MLP: two groups of 16 octonions (lanes 0-15 / 16-31) ----
        float sc[2][3];
#pragma unroll
        for (int g = 0; g < 2; ++g) {
            float rg = g ? __shfl_xor(rinv, 16) : rinv;
            float ng = g ? __shfl_xor(nrm,  16) : nrm;
            v2f bmat;                     // B = invT : VGPR0 = K=0 row (r), VGPR1 = K=1 row (n)
            bmat.x = lo ? rg : 0.0f;      // lanes>=16 are K=2,3 padding -> 0
            bmat.y = lo ? ng : 0.0f;

            float a0 = 0.0f, a1 = 0.0f, a2 = 0.0f;
#pragma unroll
            for (int t = 0; t < 2; ++t) {
                v8f d = __builtin_amdgcn_wmma_f32_16x16x4_f32(
                    false, amat[t], false, bmat, (short)0, cbias[t], false, false);
#pragma unroll
                for (int v = 0; v < 8; ++v) {
                    float h = gelu_exact(d[v]);
                    a0 = fmaf(h, w2c[0][t][v], a0);
                    a1 = fmaf(h, w2c[1][t][v], a1);
                    a2 = fmaf(h, w2c[2][t][v], a2);
                }
            }
            // combine hidden halves living on lane l and lane l^16 (same octonion)
            a0 += __shfl_xor(a0, 16);
            a1 += __shfl_xor(a1, 16);
            a2 += __shfl_xor(a2, 16);
            sc[g][0] = a0 + bb0;
            sc[g][1] = a1 + bb1;
            sc[g][2] = a2 + bb2v;
        }
        // lane l<16 owns a group-0 octonion (local idx l); lane l>=16 a group-1 one.
        const float sa  = lo ? sc[0][0] : sc[1][0];
        const float sb  = lo ? sc[0][1] : sc[1][1];
        const float scx = lo ? sc[0][2] : sc[1][2];

        // ---- octonion polynomial, normalize, blend (all per-lane) ----
        Oct o2 = omul(oh, oh);
        Oct o3 = omul(oh, o2);

        float u[8];
        u[0] = sa*oh.a.w + sb*o2.a.w + scx*o3.a.w;
        u[1] = sa*oh.a.x + sb*o2.a.x + scx*o3.a.x;
        u[2] = sa*oh.a.y + sb*o2.a.y + scx*o3.a.y;
        u[3] = sa*oh.a.z + sb*o2.a.z + scx*o3.a.z;
        u[4] = sa*oh.b.w + sb*o2.b.w + scx*o3.b.w;
        u[5] = sa*oh.b.x + sb*o2.b.x + scx*o3.b.x;
        u[6] = sa*oh.b.y + sb*o2.b.y + scx*o3.b.y;
        u[7] = sa*oh.b.z + sb*o2.b.z + scx*o3.b.z;

        float usq = 0.0f;
#pragma unroll
        for (int k = 0; k < 8; ++k) usq = fmaf(u[k], u[k], usq);
        const float unrm = __builtin_amdgcn_sqrtf(usq);
        const float rcp  = __builtin_amdgcn_rcpf(fmaxf(unrm, 1e-8f));
        const float s = lam * rcp;

        v4f r0, r1;
        r0.x = fmaf(oml, q0.x, s*u[0]);
        r0.y = fmaf(oml, q0.y, s*u[1]);
        r0.z = fmaf(oml, q0.z, s*u[2]);
        r0.w = fmaf(oml, q0.w, s*u[3]);
        r1.x = fmaf(oml, q1.x, s*u[4]);
        r1.y = fmaf(oml, q1.y, s*u[5]);
        r1.z = fmaf(oml, q1.z, s*u[6]);
        r1.w = fmaf(oml, q1.w, s*u[7]);

        v4f* wp = (v4f*)out + base;
        __builtin_nontemporal_store(r0, wp);
        __builtin_nontemporal_store(r1, wp + 1);
    }
}

extern "C" void kernel_launch(void* const* d_in, const int* in_sizes, int n_in,
                              void* d_out, int out_size, void* d_ws, size_t ws_size,
                              hipStream_t stream) {
    const float* o     = (const float*)d_in[0];
    const float* W1    = (const float*)d_in[1];
    const float* b1    = (const float*)d_in[2];
    const float* W2    = (const float*)d_in[3];
    const float* b2    = (const float*)d_in[4];
    const float* alpha = (const float*)d_in[5];
    float* out = (float*)d_out;

    const int n_oct = in_sizes[0] / 8;           // 4,194,304 for the reference shapes
    const int ITERS = 4;                         // octonions per thread (must match kernel)
    const int block = 256;                       // 8 wave32s per block
    int threads = (n_oct + ITERS - 1) / ITERS;
    int blocks  = (threads + block - 1) / block; // 4096 blocks for reference shapes
    int stride  = blocks * block;

    g2ff_kernel<<<blocks, block, 0, stream>>>(o, W1, b1, W2, b2, alpha, out,
                                              n_oct, stride);
}